// DGCNN_46514495816097
// MI455X (gfx1250) — compile-verified
//
#include <hip/hip_runtime.h>
#include <math.h>

// ---------------- DGCNN forward, CDNA5 (gfx1250) ----------------
// EdgeConv linearity trick:  [xi, xj-xi] @ W = xi@(Wt-Wb) + xj@Wb
//   -> per-point GEMM (K=20x fewer rows), then gather+max combine.
// All dense math via V_WMMA_F32_16X16X4_F32 (fp32 kept for kNN
// distance-ranking fidelity). Layer-1 features padded 3->4 so every
// K dim is a multiple of 4 (no guards). Weights are stored K-pair
// packed so every WMMA operand load is a single aligned b64.

#define BATCH 8
#define NPTS  2048
#define BNPT  (BATCH * NPTS)
#define KNN   20
#define BNEPS 1e-5f

typedef __attribute__((ext_vector_type(2))) float v2f;
typedef __attribute__((ext_vector_type(8))) float v8f;

__device__ __forceinline__ v8f wmma_f32x4(v2f a, v2f b, v8f c) {
  return __builtin_amdgcn_wmma_f32_16x16x4_f32(false, a, false, b, (short)0, c,
                                               false, false);
}

// ---------------- pad pos [BN,3] -> [BN,4] with zero 4th component -------------
__global__ void pad_pos_kernel(const float* __restrict__ pos,
                               float* __restrict__ pos4) {
  int i = blockIdx.x * blockDim.x + threadIdx.x;
  if (i >= BNPT) return;
  pos4[i * 4 + 0] = pos[i * 3 + 0];
  pos4[i * 4 + 1] = pos[i * 3 + 1];
  pos4[i * 4 + 2] = pos[i * 3 + 2];
  pos4[i * 4 + 3] = 0.f;
}

// ---------------- squared norms per point ----------------
__global__ void norms_kernel(const float* __restrict__ X, float* __restrict__ sq,
                             int F) {
  int i = blockIdx.x * blockDim.x + threadIdx.x;
  if (i >= BNPT) return;
  float s = 0.f;
  for (int f = 0; f < F; ++f) { float v = X[(size_t)i * F + f]; s += v * v; }
  sq[i] = s;
}

// ---------------- fused kNN: WMMA dot tiles + top-20 selection ----------------
// 1 wave per 16 queries; 64 candidates per chunk (4 dot sub-tiles / pass).
// dist = |q|^2 + |c|^2 - 2 q.c  via fp32 16x16x4 WMMA, F fully unrolled.
template <int F>
__global__ __launch_bounds__(32) void knn_kernel(const float* __restrict__ X,
                                                 const float* __restrict__ sq,
                                                 int* __restrict__ knn_idx) {
  __shared__ float sXq[16 * F];
  __shared__ float sDq[16];
  __shared__ float sDc[64];
  __shared__ float sDist[16 * 64];
  __shared__ float sTop[16 * KNN];
  __shared__ int   sTopI[16 * KNN];

  const int lane    = threadIdx.x;
  const int b       = blockIdx.y;
  const int qbase   = blockIdx.x * 16;
  const int rowbase = b * NPTS + qbase;

  for (int t = lane; t < 16 * F; t += 32)
    sXq[t] = X[(size_t)(rowbase + t / F) * F + (t % F)];
  if (lane < 16) sDq[lane] = sq[rowbase + lane];
  __syncthreads();

  const int arow  = lane & 15;
  const int khalf = (lane >> 4) * 2;  // lanes 0-15: K 0,1 ; lanes 16-31: K 2,3
  int cnt = 0, mp = 0;
  float mx = -1e30f;

  for (int cbase = 0; cbase < NPTS; cbase += 64) {
    sDc[lane]      = sq[b * NPTS + cbase + lane];
    sDc[lane + 32] = sq[b * NPTS + cbase + lane + 32];
    __syncthreads();

    v8f c0 = {0.f, 0.f, 0.f, 0.f, 0.f, 0.f, 0.f, 0.f};
    v8f c1 = c0, c2 = c0, c3 = c0;
    const float* Xc = X + (size_t)(b * NPTS + cbase) * F;
#pragma unroll
    for (int k0 = 0; k0 < F; k0 += 4) {
      const int ka = k0 + khalf;
      v2f a  = *(const v2f*)&sXq[arow * F + ka];
      v2f b0 = *(const v2f*)&Xc[(size_t)(arow)      * F + ka];
      v2f b1 = *(const v2f*)&Xc[(size_t)(arow + 16) * F + ka];
      v2f b2 = *(const v2f*)&Xc[(size_t)(arow + 32) * F + ka];
      v2f b3 = *(const v2f*)&Xc[(size_t)(arow + 48) * F + ka];
      c0 = wmma_f32x4(a, b0, c0);
      c1 = wmma_f32x4(a, b1, c1);
      c2 = wmma_f32x4(a, b2, c2);
      c3 = wmma_f32x4(a, b3, c3);
    }

    {  // C layout: VGPR r -> row r (lanes 0-15) / r+8 (lanes 16-31); col = lane&15
      const int m0 = (lane < 16) ? 0 : 8;
      v8f cc[4] = {c0, c1, c2, c3};
#pragma unroll
      for (int t = 0; t < 4; ++t) {
        const int col = t * 16 + arow;
#pragma unroll
        for (int r = 0; r < 8; ++r)
          sDist[(m0 + r) * 64 + col] = sDq[m0 + r] + sDc[col] - 2.f * cc[t][r];
      }
    }
    __syncthreads();

    if (lane < 16) {  // replace-max top-20 (order irrelevant: feeds a max)
      const int q = lane;
      for (int n = 0; n < 64; ++n) {
        float d = sDist[q * 64 + n];
        int  ci = b * NPTS + cbase + n;  // global row index
        if (cnt < KNN) {
          sTop[q * KNN + cnt] = d; sTopI[q * KNN + cnt] = ci;
          if (d > mx) { mx = d; mp = cnt; }
          ++cnt;
        } else if (d < mx) {
          sTop[q * KNN + mp] = d; sTopI[q * KNN + mp] = ci;
          mx = -1e30f;
          for (int t = 0; t < KNN; ++t)
            if (sTop[q * KNN + t] > mx) { mx = sTop[q * KNN + t]; mp = t; }
        }
      }
    }
    __syncthreads();
  }

  if (lane < 16)
    for (int k = 0; k < KNN; ++k)
      knn_idx[(size_t)(rowbase + lane) * KNN + k] = sTopI[lane * KNN + k];
}

// ---------------- fp32 WMMA GEMM: Y[M,Nn] = X[M,Kd] @ W[Kd,Nn] -----------------
// W is K-pair packed:  Wp[((k>>1)*Nn + n)*2 + (k&1)] = W[k][n], so every B
// operand is one aligned b64 load. One wave -> 16 rows x 64 cols (4 accums
// share the A operand). Kd %4==0, Nn %64==0, M %16==0. No guards.
__global__ __launch_bounds__(32) void gemm16x64(const float* __restrict__ X,
                                                const float* __restrict__ Wp,
                                                float* __restrict__ Y, int Kd,
                                                int Nn) {
  const int lane  = threadIdx.x;
  const int l15   = lane & 15;
  const int row   = blockIdx.y * 16 + l15;
  const int col0  = blockIdx.x * 64 + l15;
  const int khalf = (lane >> 4) * 2;

  v8f c0 = {0.f, 0.f, 0.f, 0.f, 0.f, 0.f, 0.f, 0.f};
  v8f c1 = c0, c2 = c0, c3 = c0;

  const float* xp = X + (size_t)row * Kd + khalf;
  // pair index for this lane half: ka>>1 = k0/2 + (lane>=16)
  const float* wp = Wp + ((size_t)(lane >> 4) * Nn + col0) * 2;
  for (int k0 = 0; k0 < Kd; k0 += 4) {
    __builtin_prefetch(xp + k0 + 32, 0, 3);        // near-scope prefetch
    v2f a  = *(const v2f*)(xp + k0);
    const float* w = wp + (size_t)k0 * Nn;         // (k0/2)*Nn*2
    v2f b0 = *(const v2f*)(w);
    v2f b1 = *(const v2f*)(w + 32);                // col0+16
    v2f b2 = *(const v2f*)(w + 64);                // col0+32
    v2f b3 = *(const v2f*)(w + 96);                // col0+48
    c0 = wmma_f32x4(a, b0, c0);
    c1 = wmma_f32x4(a, b1, c1);
    c2 = wmma_f32x4(a, b2, c2);
    c3 = wmma_f32x4(a, b3, c3);
  }

  const int mbase = blockIdx.y * 16 + ((lane < 16) ? 0 : 8);
  v8f cc[4] = {c0, c1, c2, c3};
#pragma unroll
  for (int t = 0; t < 4; ++t)
#pragma unroll
    for (int r = 0; r < 8; ++r)
      Y[(size_t)(mbase + r) * Nn + col0 + t * 16] = cc[t][r];
}

// --- combined weight, K-pair packed: rows = [Wtop-Wbot | Wbot], zero-padded ---
__global__ void prep_w(const float* __restrict__ W, float* __restrict__ Wp,
                       int FIN, int FINp, int F) {
  int t = blockIdx.x * blockDim.x + threadIdx.x;
  if (t >= FINp * F) return;
  int k = t / F, f = t % F;
  float wt = 0.f, wb = 0.f;
  if (k < FIN) {
    wt = W[k * F + f];
    wb = W[(FIN + k) * F + f];
  }
  const int Nn = 2 * F;
  size_t base = ((size_t)(k >> 1) * Nn) * 2 + (k & 1);
  Wp[base + (size_t)f * 2]       = wt - wb;
  Wp[base + (size_t)(F + f) * 2] = wb;
}

// ---------------- K-pair pack a plain [Kd,Nn] weight ----------------
__global__ void pack_w_kernel(const float* __restrict__ W, float* __restrict__ Wp,
                              int Kd, int Nn) {
  int t = blockIdx.x * blockDim.x + threadIdx.x;
  if (t >= Kd * Nn) return;
  int k = t / Nn, n = t % Nn;
  Wp[((size_t)(k >> 1) * Nn + n) * 2 + (k & 1)] = W[t];
}

// ---------------- out[i,f] = max_k lrelu(bn(A[i,f] + Bv[j_k,f])) ----------------
__global__ void edge_combine(const float* __restrict__ Y /*[BN,2F]: A|Bv*/,
                             const int* __restrict__ idx,
                             const float* __restrict__ g,
                             const float* __restrict__ be,
                             const float* __restrict__ mu,
                             const float* __restrict__ va,
                             float* __restrict__ out, int F) {
  int i = blockIdx.x;
  int f = threadIdx.x;
  float s = g[f] * rsqrtf(va[f] + BNEPS);
  float t = be[f] - mu[f] * s;
  float a = Y[(size_t)i * 2 * F + f];
  const int* ip = idx + (size_t)i * KNN;
  float m = -1e30f;
#pragma unroll 4
  for (int k = 0; k < KNN; ++k) {
    int j = ip[k];
    float v = (a + Y[(size_t)j * 2 * F + F + f]) * s + t;
    v = (v > 0.f) ? v : 0.2f * v;
    m = fmaxf(m, v);
  }
  out[(size_t)i * F + f] = m;
}

// ---------------- concat [x1|x2|x3|x4] -> [BN,512] ----------------
__global__ void concat_kernel(const float* __restrict__ x1,
                              const float* __restrict__ x2,
                              const float* __restrict__ x3,
                              const float* __restrict__ x4,
                              float* __restrict__ cat) {
  int i = blockIdx.x, t = threadIdx.x;
  float v;
  if (t < 64)        v = x1[(size_t)i * 64 + t];
  else if (t < 128)  v = x2[(size_t)i * 64 + (t - 64)];
  else if (t < 256)  v = x3[(size_t)i * 128 + (t - 128)];
  else               v = x4[(size_t)i * 256 + (t - 256)];
  cat[(size_t)i * 512 + t] = v;
}

// ---------------- bn+lrelu then max & mean pool over N ----------------
__global__ void pool_kernel(const float* __restrict__ hm /*[B,N,1024]*/,
                            const float* __restrict__ g,
                            const float* __restrict__ be,
                            const float* __restrict__ mu,
                            const float* __restrict__ va,
                            float* __restrict__ pooled /*[B,2048]*/) {
  int b = blockIdx.y;
  int f = blockIdx.x * blockDim.x + threadIdx.x;  // 0..1023
  float s = g[f] * rsqrtf(va[f] + BNEPS);
  float t = be[f] - mu[f] * s;
  float mx = -1e30f, sum = 0.f;
  for (int n = 0; n < NPTS; ++n) {
    float v = hm[((size_t)b * NPTS + n) * 1024 + f] * s + t;
    v = (v > 0.f) ? v : 0.2f * v;
    mx = fmaxf(mx, v);
    sum += v;
  }
  pooled[(size_t)b * 2048 + f]        = mx;
  pooled[(size_t)b * 2048 + 1024 + f] = sum * (1.f / (float)NPTS);
}

// ---------------- small FC head (M=8): dot + optional bias/bn/lrelu ------------
__global__ void fc_kernel(const float* __restrict__ X, const float* __restrict__ W,
                          const float* __restrict__ bias,
                          const float* __restrict__ g, const float* __restrict__ be,
                          const float* __restrict__ mu, const float* __restrict__ va,
                          float* __restrict__ Y, int Kd, int Nn, int act) {
  int m = blockIdx.y;
  int n = blockIdx.x * blockDim.x + threadIdx.x;
  if (n >= Nn) return;
  float acc = bias ? bias[n] : 0.f;
  for (int k = 0; k < Kd; ++k) acc += X[(size_t)m * Kd + k] * W[(size_t)k * Nn + n];
  if (act) {
    float s = g[n] * rsqrtf(va[n] + BNEPS);
    acc = acc * s + (be[n] - mu[n] * s);
    acc = (acc > 0.f) ? acc : 0.2f * acc;
  }
  Y[(size_t)m * Nn + n] = acc;
}

extern "C" void kernel_launch(void* const* d_in, const int* in_sizes, int n_in,
                              void* d_out, int out_size, void* d_ws, size_t ws_size,
                              hipStream_t stream) {
  (void)in_sizes; (void)n_in; (void)out_size; (void)ws_size;
  const float* pos   = (const float*)d_in[0];
  const float* W1    = (const float*)d_in[1];
  const float* g1    = (const float*)d_in[2];
  const float* be1   = (const float*)d_in[3];
  const float* mu1   = (const float*)d_in[4];
  const float* va1   = (const float*)d_in[5];
  const float* W2    = (const float*)d_in[6];
  const float* g2    = (const float*)d_in[7];
  const float* be2   = (const float*)d_in[8];
  const float* mu2   = (const float*)d_in[9];
  const float* va2   = (const float*)d_in[10];
  const float* W3    = (const float*)d_in[11];
  const float* g3    = (const float*)d_in[12];
  const float* be3   = (const float*)d_in[13];
  const float* mu3   = (const float*)d_in[14];
  const float* va3   = (const float*)d_in[15];
  const float* W4    = (const float*)d_in[16];
  const float* g4    = (const float*)d_in[17];
  const float* be4   = (const float*)d_in[18];
  const float* mu4   = (const float*)d_in[19];
  const float* va4   = (const float*)d_in[20];
  const float* Wm    = (const float*)d_in[21];
  const float* gm    = (const float*)d_in[22];
  const float* bem   = (const float*)d_in[23];
  const float* mum   = (const float*)d_in[24];
  const float* vam   = (const float*)d_in[25];
  const float* Wa    = (const float*)d_in[26];
  const float* ga    = (const float*)d_in[27];
  const float* bea   = (const float*)d_in[28];
  const float* mua   = (const float*)d_in[29];
  const float* vaa   = (const float*)d_in[30];
  const float* Wb    = (const float*)d_in[31];
  const float* gb    = (const float*)d_in[32];
  const float* beb   = (const float*)d_in[33];
  const float* mub   = (const float*)d_in[34];
  const float* vab   = (const float*)d_in[35];
  const float* biasb = (const float*)d_in[36];
  const float* Wcls  = (const float*)d_in[37];
  const float* biasc = (const float*)d_in[38];

  char* ws = (char*)d_ws;
  size_t off = 0;
  auto alloc = [&](size_t bytes) -> float* {
    float* p = (float*)(ws + off);
    off += (bytes + 255) & ~(size_t)255;
    return p;
  };
  float* pos4   = alloc((size_t)BNPT * 4 * 4);
  float* sq     = alloc((size_t)BNPT * 4);
  int*   idxb   = (int*)alloc((size_t)BNPT * KNN * 4);
  float* Wcmb   = alloc((size_t)256 * 512 * 4);   // max FINp x 2*FOUT (packed)
  float* Wmp    = alloc((size_t)512 * 1024 * 4);  // packed Wm
  float* Yab    = alloc((size_t)BNPT * 512 * 4);  // [A | Bv]; reused as cat
  float* x1     = alloc((size_t)BNPT * 64 * 4);
  float* x2     = alloc((size_t)BNPT * 64 * 4);
  float* x3     = alloc((size_t)BNPT * 128 * 4);
  float* x4     = alloc((size_t)BNPT * 256 * 4);
  float* hm     = alloc((size_t)BNPT * 1024 * 4);
  float* pooled = alloc((size_t)BATCH * 2048 * 4);
  float* h1     = alloc((size_t)BATCH * 512 * 4);
  float* h2     = alloc((size_t)BATCH * 256 * 4);
  float* cat    = Yab;  // Yab free after layer 4's combine

  pad_pos_kernel<<<dim3((BNPT + 255) / 256), 256, 0, stream>>>(pos, pos4);

  auto edge_layer = [&](const float* Xin, int FIN, int FINp, int FOUT,
                        const float* W, const float* g, const float* be,
                        const float* mu, const float* va, float* Xout) {
    norms_kernel<<<dim3((BNPT + 255) / 256), 256, 0, stream>>>(Xin, sq, FINp);
    dim3 kg(NPTS / 16, BATCH);
    if (FINp == 4)
      knn_kernel<4><<<kg, 32, 0, stream>>>(Xin, sq, idxb);
    else if (FINp == 64)
      knn_kernel<64><<<kg, 32, 0, stream>>>(Xin, sq, idxb);
    else
      knn_kernel<128><<<kg, 32, 0, stream>>>(Xin, sq, idxb);
    int nw = FINp * FOUT;
    prep_w<<<dim3((nw + 255) / 256), 256, 0, stream>>>(W, Wcmb, FIN, FINp, FOUT);
    gemm16x64<<<dim3((2 * FOUT) / 64, BNPT / 16), 32, 0, stream>>>(
        Xin, Wcmb, Yab, FINp, 2 * FOUT);
    edge_combine<<<dim3(BNPT), FOUT, 0, stream>>>(Yab, idxb, g, be, mu, va, Xout,
                                                  FOUT);
  };

  edge_layer(pos4, 3,   4,   64,  W1, g1, be1, mu1, va1, x1);
  edge_layer(x1,   64,  64,  64,  W2, g2, be2, mu2, va2, x2);
  edge_layer(x2,   64,  64,  128, W3, g3, be3, mu3, va3, x3);
  edge_layer(x3,   128, 128, 256, W4, g4, be4, mu4, va4, x4);

  concat_kernel<<<dim3(BNPT), 512, 0, stream>>>(x1, x2, x3, x4, cat);
  pack_w_kernel<<<dim3((512 * 1024 + 255) / 256), 256, 0, stream>>>(Wm, Wmp, 512,
                                                                    1024);
  gemm16x64<<<dim3(1024 / 64, BNPT / 16), 32, 0, stream>>>(cat, Wmp, hm, 512,
                                                           1024);
  pool_kernel<<<dim3(4, BATCH), 256, 0, stream>>>(hm, gm, bem, mum, vam, pooled);

  fc_kernel<<<dim3(2, BATCH), 256, 0, stream>>>(pooled, Wa, nullptr, ga, bea, mua,
                                                vaa, h1, 2048, 512, 1);
  fc_kernel<<<dim3(1, BATCH), 256, 0, stream>>>(h1, Wb, biasb, gb, beb, mub, vab,
                                                h2, 512, 256, 1);
  fc_kernel<<<dim3(1, BATCH), 256, 0, stream>>>(h2, Wcls, biasc, nullptr, nullptr,
                                                nullptr, nullptr, (float*)d_out,
                                                256, 40, 0);
}